// MultiBoxLoss_combined_56453050138984
// MI455X (gfx1250) — compile-verified
//
#include <hip/hip_runtime.h>
#include <math.h>

// MultiBoxLoss (few-shot SSD) for MI455X / gfx1250 (wave32).
// Memory-bound (~250 MB traffic -> ~11us @ 23.3 TB/s); single streaming pass
// per tensor. The one GEMM (qn @ proto^T, [581k,64]x[64,5]) uses
// V_WMMA_F32_16X16X4_F32 (exact f32), K=64 as 16 k-steps.
// B fragments are loaded branchlessly (clamped index * mask) so the 16-WMMA
// chain issues without per-step exec-mask toggles / dscnt stalls.
//
// Workspace layout (floats/ints, ~34 MB total; assumed <= ws_size).

#define PRI 11620
#define NIMG 75        // 25 support + 50 query images
#define NSI 25
#define NQI 50
#define NWAY 5
#define NOBJ 8
#define CCH 64

typedef __attribute__((ext_vector_type(2))) float v2f;
typedef __attribute__((ext_vector_type(8))) float v8f;

// ---------------------------------------------------------------- matching
__global__ void fsl_match_kernel(const float* __restrict__ priors,
                                 const float* __restrict__ s_t,
                                 const float* __restrict__ q_t,
                                 float* __restrict__ btv, int* __restrict__ bti,
                                 float* __restrict__ loc_t, int* __restrict__ conf_t,
                                 int* __restrict__ obj_t, int* __restrict__ num_pos) {
  const int img = blockIdx.x;          // 0..74 (s first, then q)
  const int tid = threadIdx.x;         // 256 threads
  __shared__ float tbx[NOBJ][4];
  __shared__ float lab[NOBJ];
  __shared__ float sv[NOBJ][256];
  __shared__ int   si[NOBJ][256];
  __shared__ int   bp[NOBJ];
  __shared__ int   scnt[256];

  if (tid < NOBJ) {
    const float* t5;
    float lv;
    if (img < NSI) {
      t5 = s_t + ((size_t)img * NOBJ + tid) * 5;
      lv = t5[4];
    } else {
      int qi = img - NSI;
      t5 = q_t + ((size_t)qi * NOBJ + tid) * 5;
      int way = qi / 10;
      lv = (t5[4] == -1.0f) ? -1.0f : (float)(way + 1);
    }
    tbx[tid][0] = t5[0]; tbx[tid][1] = t5[1];
    tbx[tid][2] = t5[2]; tbx[tid][3] = t5[3];
    lab[tid] = lv;
  }
  __syncthreads();

  float bv[NOBJ]; int bi[NOBJ];
  #pragma unroll
  for (int t = 0; t < NOBJ; ++t) { bv[t] = -1.0f; bi[t] = 0; }

  // pass 1: IoU, per-prior best truth (first-max), per-truth best prior (stripe-local)
  for (int p = tid; p < PRI; p += 256) {
    float cx = priors[p * 4 + 0], cy = priors[p * 4 + 1];
    float w  = priors[p * 4 + 2], h  = priors[p * 4 + 3];
    float px1 = cx - w * 0.5f, py1 = cy - h * 0.5f;
    float px2 = cx + w * 0.5f, py2 = cy + h * 0.5f;
    float pa = (px2 - px1) * (py2 - py1);
    float bestov = -1.0f; int bestt = 0;
    #pragma unroll
    for (int t = 0; t < NOBJ; ++t) {
      float x1 = tbx[t][0], y1 = tbx[t][1], x2 = tbx[t][2], y2 = tbx[t][3];
      float ix = fminf(x2, px2) - fmaxf(x1, px1);
      float iy = fminf(y2, py2) - fmaxf(y1, py1);
      ix = fmaxf(ix, 0.0f); iy = fmaxf(iy, 0.0f);
      float inter = ix * iy;
      float ta = (x2 - x1) * (y2 - y1);
      float iou = inter / (ta + pa - inter);
      if (iou > bestov) { bestov = iou; bestt = t; }   // first t wins on ties
      if (iou > bv[t])  { bv[t] = iou; bi[t] = p; }    // first p wins on ties
    }
    btv[(size_t)img * PRI + p] = bestov;
    bti[(size_t)img * PRI + p] = bestt;
  }
  #pragma unroll
  for (int t = 0; t < NOBJ; ++t) { sv[t][tid] = bv[t]; si[t][tid] = bi[t]; }
  __syncthreads();
  if (tid < NOBJ) {
    float best = -2.0f; int idx = 0x7fffffff;
    for (int j = 0; j < 256; ++j) {
      float v = sv[tid][j]; int ii = si[tid][j];
      if (v > best || (v == best && ii < idx)) { best = v; idx = ii; }
    }
    bp[tid] = idx;
  }
  __syncthreads();

  // pass 2: overrides, conf/loc/obj
  int cnt = 0;
  for (int p = tid; p < PRI; p += 256) {
    float ov = btv[(size_t)img * PRI + p];
    int ti   = bti[(size_t)img * PRI + p];
    #pragma unroll
    for (int t = 0; t < NOBJ; ++t)
      if (bp[t] == p) { ov = 2.0f; ti = t; }           // last t wins
    float lv = lab[ti];
    float conf = (ov < 0.5f) ? 0.0f : lv;
    int ci = (int)conf;
    int pos = (conf > 0.0f) ? 1 : 0;
    float x1 = tbx[ti][0], y1 = tbx[ti][1], x2 = tbx[ti][2], y2 = tbx[ti][3];
    float cx = priors[p * 4 + 0], cy = priors[p * 4 + 1];
    float w  = priors[p * 4 + 2], h  = priors[p * 4 + 3];
    size_t o = (size_t)img * PRI + p;
    loc_t[o * 4 + 0] = ((x1 + x2) * 0.5f - cx) / (0.1f * w);
    loc_t[o * 4 + 1] = ((y1 + y2) * 0.5f - cy) / (0.1f * h);
    loc_t[o * 4 + 2] = logf((x2 - x1) / w) / 0.2f;
    loc_t[o * 4 + 3] = logf((y2 - y1) / h) / 0.2f;
    conf_t[o] = ci;
    obj_t[o] = pos;
    cnt += pos;
  }
  scnt[tid] = cnt; __syncthreads();
  for (int s = 128; s > 0; s >>= 1) {
    if (tid < s) scnt[tid] += scnt[tid + s];
    __syncthreads();
  }
  if (tid == 0) num_pos[img] = scnt[0];
}

// ---------------------------------------------------------------- prototypes
__global__ void fsl_proto_kernel(const float* __restrict__ s_conf,
                                 const int* __restrict__ obj_t,
                                 float* __restrict__ proto) {
  const int way = blockIdx.x;          // 0..4
  const int tid = threadIdx.x;         // 128 threads = 4 waves
  const int lane = tid & 31, wv = tid >> 5;
  float acc[CCH];
  #pragma unroll
  for (int c = 0; c < CCH; ++c) acc[c] = 0.0f;
  int cnt = 0;
  for (int sp = tid; sp < 5 * PRI; sp += 128) {
    int sh = sp / PRI, p = sp - sh * PRI;
    int img = way * 5 + sh;
    if (obj_t[(size_t)img * PRI + p]) {
      const float* v = s_conf + ((size_t)img * PRI + p) * CCH;
      float ss = 0.0f;
      #pragma unroll
      for (int c = 0; c < CCH; ++c) { float x = fmaxf(v[c], 0.0f); ss += x * x; }
      float inv = 1.0f / sqrtf(ss);
      #pragma unroll
      for (int c = 0; c < CCH; ++c) acc[c] += fmaxf(v[c], 0.0f) * inv;
      ++cnt;
    }
  }
  __shared__ float sacc[4][CCH];
  __shared__ int   scn[4];
  #pragma unroll
  for (int c = 0; c < CCH; ++c) {
    float v = acc[c];
    v += __shfl_down(v, 16); v += __shfl_down(v, 8);
    v += __shfl_down(v, 4);  v += __shfl_down(v, 2); v += __shfl_down(v, 1);
    if (lane == 0) sacc[wv][c] = v;
  }
  {
    int c2 = cnt;
    c2 += __shfl_down(c2, 16); c2 += __shfl_down(c2, 8);
    c2 += __shfl_down(c2, 4);  c2 += __shfl_down(c2, 2); c2 += __shfl_down(c2, 1);
    if (lane == 0) scn[wv] = c2;
  }
  __syncthreads();
  __shared__ float pr[CCH];
  __shared__ float invn;
  if (tid < CCH) {
    float t = sacc[0][tid] + sacc[1][tid] + sacc[2][tid] + sacc[3][tid];
    int ct = scn[0] + scn[1] + scn[2] + scn[3];
    pr[tid] = t / (float)ct;
  }
  __syncthreads();
  if (tid == 0) {
    float ss = 0.0f;
    for (int c = 0; c < CCH; ++c) ss += pr[c] * pr[c];
    invn = 1.0f / sqrtf(ss);
  }
  __syncthreads();
  if (tid < CCH) proto[way * CCH + tid] = pr[tid] * invn;
}

// ------------------------------------------------- query class-CE (WMMA f32)
__global__ void __launch_bounds__(32)
fsl_qce_wmma_kernel(const float* __restrict__ q_conf,
                    const float* __restrict__ q_obj,
                    const float* __restrict__ proto,
                    const int* __restrict__ conf_t,
                    float* __restrict__ ce_c) {
  const int tile = blockIdx.x;   // 16-prior tile
  const int q    = blockIdx.y;   // 0..49
  const int lane = threadIdx.x;  // 32 (one wave)
  const int n = lane & 15, h = lane >> 4;
  __shared__ float sproto[NWAY * CCH];
  for (int i = lane; i < NWAY * CCH; i += 32) sproto[i] = proto[i];
  __syncthreads();

  const int p0 = tile * 16;
  int prow = p0 + n; if (prow > PRI - 1) prow = PRI - 1;
  const float* row = q_conf + ((size_t)q * PRI + prow) * CCH;

  // A fragments: 16x4 f32 per k-step; lane<16 holds K=4s,4s+1; lane>=16 K=4s+2,4s+3
  v2f a[16];
  float ss = 0.0f;
  #pragma unroll
  for (int s = 0; s < 16; ++s) {
    int c = 4 * s + 2 * h;
    float x = row[c], y = row[c + 1];
    a[s].x = x; a[s].y = y;
    ss += x * x + y * y;
  }
  ss += __shfl_xor(ss, 16);            // combine the two half-rows
  float inv = 1.0f / sqrtf(ss);
  #pragma unroll
  for (int s = 0; s < 16; ++s) { a[s].x *= inv; a[s].y *= inv; }

  // B fragments: 4x16 per k-step; column n holds proto[n][.] for n<5 else 0.
  // Branchless (clamped index * mask) so EXEC stays full and all ds_loads can
  // be issued/waited once before the back-to-back WMMA chain.
  const int   np  = (n < NWAY) ? n : 0;
  const float msk = (n < NWAY) ? 1.0f : 0.0f;
  const float* bp = sproto + np * CCH + 2 * h;
  v2f b[16];
  #pragma unroll
  for (int s = 0; s < 16; ++s) {
    b[s].x = bp[4 * s]     * msk;
    b[s].y = bp[4 * s + 1] * msk;
  }

  v8f dacc = {0.f, 0.f, 0.f, 0.f, 0.f, 0.f, 0.f, 0.f};
  #pragma unroll
  for (int s = 0; s < 16; ++s) {
    dacc = __builtin_amdgcn_wmma_f32_16x16x4_f32(false, a[s], false, b[s],
                                                 (short)0, dacc, false, false);
  }

  __shared__ float td[16][16];
  #pragma unroll
  for (int i = 0; i < 8; ++i) td[i + 8 * h][n] = dacc[i];
  __syncthreads();

  if (lane < 16) {
    int p = p0 + lane;
    if (p < PRI) {
      float c5[NWAY];
      float es = 0.0f;
      #pragma unroll
      for (int k = 0; k < NWAY; ++k) { c5[k] = 20.0f * td[lane][k]; es += expf(c5[k]); }
      const float* o = q_obj + ((size_t)q * PRI + p) * 2;
      float l[NWAY + 1];
      l[0] = o[0] + logf(es);
      #pragma unroll
      for (int k = 0; k < NWAY; ++k) l[k + 1] = o[1] + c5[k];
      float m = l[0];
      #pragma unroll
      for (int k = 1; k <= NWAY; ++k) m = fmaxf(m, l[k]);
      float se = 0.0f;
      #pragma unroll
      for (int k = 0; k <= NWAY; ++k) se += expf(l[k] - m);
      float lse = m + logf(se);
      int ct = conf_t[((size_t)(NSI + q)) * PRI + p];
      int tgt = ct > 0 ? ct : 0;
      ce_c[(size_t)q * PRI + p] = lse - l[tgt];
    }
  }
}

// --------------------------------------------------------------- obj CE
__global__ void fsl_objce_kernel(const float* __restrict__ s_obj,
                                 const float* __restrict__ q_obj,
                                 const int* __restrict__ obj_t,
                                 float* __restrict__ ce_o) {
  size_t idx = (size_t)blockIdx.x * blockDim.x + threadIdx.x;
  if (idx >= (size_t)NIMG * PRI) return;
  int img = (int)(idx / PRI);
  int p = (int)(idx - (size_t)img * PRI);
  const float* o = (img < NSI) ? s_obj + ((size_t)img * PRI + p) * 2
                               : q_obj + ((size_t)(img - NSI) * PRI + p) * 2;
  float o0 = o[0], o1 = o[1];
  float m = fmaxf(o0, o1);
  float lse = m + logf(expf(o0 - m) + expf(o1 - m));
  int t = obj_t[idx];
  ce_o[idx] = lse - (t ? o1 : o0);
}

// ------------------------------- hard-negative mining + masked loss sum
// Exact top-k of ce_mine (descending, ties -> smaller index), via 1024-bin
// float-bits histogram radix select + exact rank inside the threshold bin.
__global__ void fsl_mine_kernel(const float* __restrict__ ce,
                                const int* __restrict__ obj_t,
                                const int* __restrict__ conf_t,
                                const int* __restrict__ num_pos,
                                float* __restrict__ partials,
                                int img_off, int is_class) {
  const int li = blockIdx.x;           // local image index into ce
  const int gi = li + img_off;         // global image index
  const float* cei = ce + (size_t)li * PRI;
  const int* objp = obj_t + (size_t)gi * PRI;
  const int tid = threadIdx.x;         // 256
  __shared__ int hist[1024];
  __shared__ int tb_s, rem_s;
  for (int i = tid; i < 1024; i += 256) hist[i] = 0;
  __syncthreads();
  for (int p = tid; p < PRI; p += 256) {
    unsigned key = objp[p] ? 0u : __float_as_uint(fmaxf(cei[p], 0.0f));
    atomicAdd(&hist[key >> 22], 1);
  }
  __syncthreads();
  if (tid == 0) {
    int npos = num_pos[gi];
    int nn = 3 * npos; if (nn > PRI - 1) nn = PRI - 1;
    if (nn <= 0) { tb_s = 1024; rem_s = 0; }
    else {
      int cum = 0, b = 0;
      for (b = 1023; b >= 0; --b) { cum += hist[b]; if (cum >= nn) break; }
      tb_s = b;
      rem_s = nn - (cum - hist[b]);
    }
  }
  __syncthreads();
  const int tb = tb_s, rem = rem_s;
  const int htb = (tb < 1024) ? hist[tb] : 0;
  float local = 0.0f;
  for (int p = tid; p < PRI; p += 256) {
    unsigned key = objp[p] ? 0u : __float_as_uint(fmaxf(cei[p], 0.0f));
    int bin = (int)(key >> 22);
    bool neg = false;
    if (tb < 1024) {
      if (bin > tb) neg = true;
      else if (bin == tb && rem > 0) {
        if (rem >= htb) neg = true;
        else {
          int rank = 0;
          for (int j = 0; j < PRI; ++j) {
            unsigned kj = objp[j] ? 0u : __float_as_uint(fmaxf(cei[j], 0.0f));
            if ((int)(kj >> 22) == tb && (kj > key || (kj == key && j < p))) ++rank;
          }
          neg = rank < rem;
        }
      }
    }
    bool pos = objp[p] != 0;
    bool sel;
    if (is_class) sel = (pos || neg) && (conf_t[(size_t)gi * PRI + p] >= 0);
    else          sel = pos || neg;
    if (sel) local += cei[p];
  }
  __shared__ float red[256];
  red[tid] = local; __syncthreads();
  for (int s = 128; s > 0; s >>= 1) {
    if (tid < s) red[tid] += red[tid + s];
    __syncthreads();
  }
  if (tid == 0) partials[li] = red[0];
}

// --------------------------------------------------------------- smooth-L1
__global__ void fsl_lossl_kernel(const float* __restrict__ s_loc,
                                 const float* __restrict__ q_loc,
                                 const float* __restrict__ loc_t,
                                 const int* __restrict__ obj_t,
                                 float* __restrict__ part_l) {
  const int img = blockIdx.x;
  const int tid = threadIdx.x;
  const float* ld = (img < NSI) ? s_loc + (size_t)img * PRI * 4
                                : q_loc + (size_t)(img - NSI) * PRI * 4;
  float local = 0.0f;
  for (int p = tid; p < PRI; p += 256) {
    if (obj_t[(size_t)img * PRI + p]) {
      #pragma unroll
      for (int k = 0; k < 4; ++k) {
        float d = fabsf(ld[(size_t)p * 4 + k] - loc_t[((size_t)img * PRI + p) * 4 + k]);
        local += (d < 1.0f) ? 0.5f * d * d : d - 0.5f;
      }
    }
  }
  __shared__ float red[256];
  red[tid] = local; __syncthreads();
  for (int s = 128; s > 0; s >>= 1) {
    if (tid < s) red[tid] += red[tid + s];
    __syncthreads();
  }
  if (tid == 0) part_l[img] = red[0];
}

// --------------------------------------------------------------- finalize
__global__ void fsl_final_kernel(const float* __restrict__ part_l,
                                 const float* __restrict__ part_c,
                                 const float* __restrict__ part_o,
                                 const int* __restrict__ num_pos,
                                 float* __restrict__ out) {
  if (threadIdx.x == 0 && blockIdx.x == 0) {
    float ll = 0.0f, lc = 0.0f, lo = 0.0f;
    for (int i = 0; i < NIMG; ++i) ll += part_l[i];
    for (int i = 0; i < NQI; ++i) lc += part_c[i];
    for (int i = 0; i < NIMG; ++i) lo += part_o[i];
    int sqN = 0, qN = 0;
    for (int i = 0; i < NIMG; ++i) sqN += num_pos[i];
    for (int i = NSI; i < NIMG; ++i) qN += num_pos[i];
    out[0] = ll / (float)sqN;
    out[1] = lc / (float)qN;
    out[2] = lo / (float)sqN;
  }
}

// ---------------------------------------------------------------- launch
extern "C" void kernel_launch(void* const* d_in, const int* in_sizes, int n_in,
                              void* d_out, int out_size, void* d_ws, size_t ws_size,
                              hipStream_t stream) {
  (void)in_sizes; (void)n_in; (void)out_size; (void)ws_size;
  const float* s_loc  = (const float*)d_in[0];
  const float* s_conf = (const float*)d_in[1];
  const float* s_obj  = (const float*)d_in[2];
  const float* q_loc  = (const float*)d_in[3];
  const float* q_conf = (const float*)d_in[4];
  const float* q_obj  = (const float*)d_in[5];
  const float* priors = (const float*)d_in[6];
  const float* s_t    = (const float*)d_in[7];
  const float* q_t    = (const float*)d_in[8];
  float* out = (float*)d_out;

  const size_t NP = (size_t)NIMG * PRI;      // 871500
  float* w      = (float*)d_ws;
  float* btv    = w;                         // NP
  int*   bti    = (int*)(w + NP);            // NP
  float* loc_t  = w + 2 * NP;                // 4*NP
  int*   conf_t = (int*)(w + 6 * NP);        // NP
  int*   obj_t  = (int*)(w + 7 * NP);        // NP
  float* ce_c   = w + 8 * NP;                // NQI*PRI
  float* ce_o   = ce_c + (size_t)NQI * PRI;  // NP
  float* proto  = ce_o + NP;                 // 5*64
  float* part_l = proto + NWAY * CCH;        // NIMG
  float* part_c = part_l + NIMG;             // NQI
  float* part_o = part_c + NQI;              // NIMG
  int*   num_pos = (int*)(part_o + NIMG);    // NIMG

  fsl_match_kernel<<<NIMG, 256, 0, stream>>>(priors, s_t, q_t, btv, bti,
                                             loc_t, conf_t, obj_t, num_pos);
  fsl_proto_kernel<<<NWAY, 128, 0, stream>>>(s_conf, obj_t, proto);
  {
    dim3 g((PRI + 15) / 16, NQI);
    fsl_qce_wmma_kernel<<<g, 32, 0, stream>>>(q_conf, q_obj, proto, conf_t, ce_c);
  }
  {
    size_t tot = (size_t)NIMG * PRI;
    int blocks = (int)((tot + 255) / 256);
    fsl_objce_kernel<<<blocks, 256, 0, stream>>>(s_obj, q_obj, obj_t, ce_o);
  }
  fsl_mine_kernel<<<NQI, 256, 0, stream>>>(ce_c, obj_t, conf_t, num_pos, part_c, NSI, 1);
  fsl_mine_kernel<<<NIMG, 256, 0, stream>>>(ce_o, obj_t, conf_t, num_pos, part_o, 0, 0);
  fsl_lossl_kernel<<<NIMG, 256, 0, stream>>>(s_loc, q_loc, loc_t, obj_t, part_l);
  fsl_final_kernel<<<1, 32, 0, stream>>>(part_l, part_c, part_o, num_pos, out);
}